// GNN_85383949845018
// MI455X (gfx1250) — compile-verified
//
#include <hip/hip_runtime.h>
#include <cstdint>
#include <cmath>

typedef __attribute__((ext_vector_type(2))) float v2f;
typedef __attribute__((ext_vector_type(8))) float v8f;

#define RREL 8

// ---------------------------------------------------------------------------
// Batched per-relation GEMM:  XW[r] = X @ W[r]   (fp32, V_WMMA_F32_16X16X4_F32)
//
// grid = (N/80, 1, R), block = (32, 4). Each wave owns one 16-wide column tile
// of H=64 and caches ALL its B fragments (K/4 x v2f) in registers, sweeping
// them across 5 row sub-tiles. The 80xK A tile is staged global->LDS with
// CDNA5 async loads (ASYNCcnt path), then fragments come from ds_load.
// ---------------------------------------------------------------------------
template <int K>
__global__ void rgat_gemm_wmma(const float* __restrict__ X,   // [N, K]
                               const float* __restrict__ Wt,  // [R, K, H]
                               float* __restrict__ XW,        // [R, N, H]
                               int n, int H)
{
    constexpr int ROWS = 80;                    // 100000 % 80 == 0
    __shared__ float As[ROWS * K];              // 40 KB (K=128) / 20 KB (K=64)
    const int rowBase = blockIdx.x * ROWS;
    const int r       = blockIdx.z;
    const int lane    = threadIdx.x;            // 0..31
    const int waveId  = threadIdx.y;            // 0..3
    const int tid     = waveId * 32 + lane;

    // --- async copy of ROWS contiguous rows of X into LDS (16B per lane op) ---
    const char*    gsrc  = (const char*)(X + (size_t)rowBase * K);
    const unsigned lbase = (unsigned)(uintptr_t)(void*)As;
    for (int i4 = tid; i4 < (ROWS * K) / 4; i4 += 128) {
        const unsigned loff = lbase + (unsigned)i4 * 16u;
        const char*    ga   = gsrc + (size_t)i4 * 16;
        asm volatile("global_load_async_to_lds_b128 %0, %1, off"
                     :: "v"(loff), "v"(ga) : "memory");
    }
    asm volatile("s_wait_asynccnt 0x0" ::: "memory");
    __syncthreads();

    const int half    = lane >> 4;              // 0: lanes 0-15, 1: lanes 16-31
    const int l16     = lane & 15;
    const int colBase = waveId * 16;
    const float* Wr   = Wt + (size_t)r * K * H;

    // --- preload B fragments for the entire K depth into registers ---
    // B 4x16 fp32 layout: VGPR0 = row K(2*half), VGPR1 = row K(2*half+1)
    v2f bb[K / 4];
#pragma unroll
    for (int kb4 = 0; kb4 < K / 4; ++kb4) {
        const int ka = kb4 * 4 + 2 * half;
        bb[kb4].x = Wr[(size_t)ka * H + colBase + l16];
        bb[kb4].y = Wr[(size_t)(ka + 1) * H + colBase + l16];
    }

    // --- 5 row sub-tiles reuse the register-cached B fragments ---
    for (int sub = 0; sub < ROWS / 16; ++sub) {
        const float* At = As + sub * 16 * K;
        v8f c = {0.f, 0.f, 0.f, 0.f, 0.f, 0.f, 0.f, 0.f};
#pragma unroll
        for (int kb4 = 0; kb4 < K / 4; ++kb4) {
            const int ka = kb4 * 4 + 2 * half;
            v2f a;
            // A 16x4 fp32 layout: VGPR0 = K(2*half), VGPR1 = K(2*half+1)
            a.x = At[l16 * K + ka];
            a.y = At[l16 * K + ka + 1];
            c = __builtin_amdgcn_wmma_f32_16x16x4_f32(false, a, false, bb[kb4],
                                                      (short)0, c, false, false);
        }
        // D 16x16 f32 layout: VGPR i -> row (i + 8*half), col = lane%16
        float* Dst = XW + ((size_t)r * n + rowBase + sub * 16) * H;
#pragma unroll
        for (int i = 0; i < 8; ++i)
            Dst[(size_t)(i + 8 * half) * H + colBase + l16] = c[i];
    }
}

// ---------------------------------------------------------------------------
// wq[r] = W[r] @ q ; wk[r] = W[r] @ k     (tiny: R*K*H MACs)
// ---------------------------------------------------------------------------
__global__ void compute_wqk(const float* __restrict__ Wt,  // [R,K,H]
                            const float* __restrict__ q,   // [H]
                            const float* __restrict__ k,   // [H]
                            float* __restrict__ wq, float* __restrict__ wk,
                            int K, int H)
{
    const int r = blockIdx.x;
    for (int i = threadIdx.x; i < K; i += blockDim.x) {
        const float* w = Wt + ((size_t)r * K + i) * H;
        float sq = 0.f, sk = 0.f;
        for (int h = 0; h < H; ++h) { sq += w[h] * q[h]; sk += w[h] * k[h]; }
        wq[r * K + i] = sq;
        wk[r * K + i] = sk;
    }
}

// ---------------------------------------------------------------------------
// qi[r,n] = x[n] . wq[r] ; kj[r,n] = x[n] . wk[r]   (avoids re-reading XW)
// ---------------------------------------------------------------------------
__global__ void node_qk(const float* __restrict__ X, const float* __restrict__ wq,
                        const float* __restrict__ wk, float* __restrict__ qi,
                        float* __restrict__ kj, int n, int K)
{
    const int node = blockIdx.x * blockDim.x + threadIdx.x;
    if (node >= n) return;
    float accq[RREL] = {}, acck[RREL] = {};
    const float* xr = X + (size_t)node * K;
    for (int kk = 0; kk < K; ++kk) {
        const float xv = xr[kk];
#pragma unroll
        for (int r = 0; r < RREL; ++r) {
            accq[r] += xv * wq[r * K + kk];
            acck[r] += xv * wk[r * K + kk];
        }
    }
#pragma unroll
    for (int r = 0; r < RREL; ++r) {
        qi[(size_t)r * n + node] = accq[r];
        kj[(size_t)r * n + node] = acck[r];
    }
}

// ---------------------------------------------------------------------------
// Utility fills
// ---------------------------------------------------------------------------
__global__ void fill_f32(float* __restrict__ p, float v, int count)
{
    const int i = blockIdx.x * blockDim.x + threadIdx.x;
    if (i < count) p[i] = v;
}

__global__ void fill_bias(float* __restrict__ out, const float* __restrict__ b,
                          int count, int H)
{
    const int i = blockIdx.x * blockDim.x + threadIdx.x;
    if (i < count) out[i] = b[i % H];
}

__global__ void relu_k(float* __restrict__ p, int count)
{
    const int i = blockIdx.x * blockDim.x + threadIdx.x;
    if (i < count) { const float v = p[i]; p[i] = v > 0.f ? v : 0.f; }
}

// ---------------------------------------------------------------------------
// Edge pass 1: alpha = leaky_relu(qi[t,dst] + kj[t,src], 0.2); segment max
// ---------------------------------------------------------------------------
__device__ inline void atomicMaxFloat(float* addr, float val)
{
    if (val >= 0.f) atomicMax((int*)addr, __float_as_int(val));
    else            atomicMin((unsigned int*)addr, __float_as_uint(val));
}

__global__ void edge_alpha(const int* __restrict__ src, const int* __restrict__ dst,
                           const int* __restrict__ et,
                           const float* __restrict__ qi, const float* __restrict__ kj,
                           float* __restrict__ alpha, float* __restrict__ m,
                           int E, int n)
{
    const int e = blockIdx.x * blockDim.x + threadIdx.x;
    if (e >= E) return;
    const int t = et[e], s = src[e], d = dst[e];
    float a = qi[(size_t)t * n + d] + kj[(size_t)t * n + s];
    a = a > 0.f ? a : 0.2f * a;
    alpha[e] = a;
    atomicMaxFloat(&m[d], a);
}

// ---------------------------------------------------------------------------
// Edge pass 2: ee = exp(alpha - m[dst]) (in place); segment sum -> denom
// ---------------------------------------------------------------------------
__global__ void edge_exp(const int* __restrict__ dst, float* __restrict__ alpha,
                         const float* __restrict__ m, float* __restrict__ denom, int E)
{
    const int e = blockIdx.x * blockDim.x + threadIdx.x;
    if (e >= E) return;
    const int d  = dst[e];
    const float ee = expf(alpha[e] - m[d]);
    alpha[e] = ee;
    atomicAdd(&denom[d], ee);
}

// ---------------------------------------------------------------------------
// Edge pass 3: out[dst] += (ee/denom[dst]) * XW[t, src, :]
// block = 256 -> 4 edges x 64 feature lanes; per-edge scalars via LDS broadcast
// ---------------------------------------------------------------------------
__global__ void edge_scatter(const int* __restrict__ src, const int* __restrict__ dst,
                             const int* __restrict__ et, const float* __restrict__ ee,
                             const float* __restrict__ denom, const float* __restrict__ XW,
                             float* __restrict__ out, int E, int n, int H)
{
    __shared__ int   sS[4], sD[4], sT[4];
    __shared__ float sA[4];
    const int grp = threadIdx.x >> 6;   // 0..3
    const int h   = threadIdx.x & 63;
    const int e   = blockIdx.x * 4 + grp;
    if (h == 0 && e < E) {
        const int d = dst[e];
        sS[grp] = src[e];
        sD[grp] = d;
        sT[grp] = et[e];
        sA[grp] = ee[e] / (denom[d] + 1e-16f);
    }
    __syncthreads();
    if (e < E) {
        const float v = sA[grp] * XW[((size_t)sT[grp] * n + sS[grp]) * H + h];
        atomicAdd(&out[(size_t)sD[grp] * H + h], v);
    }
}

// ---------------------------------------------------------------------------
// BatchNorm: per-channel sum / sumsq reduction, then normalize + leaky(0.01)
// blockDim=256, grid-stride multiple of 64 -> fixed channel per thread
// ---------------------------------------------------------------------------
__global__ void bn_stats(const float* __restrict__ x, float* __restrict__ stats, int total)
{
    const int idx    = blockIdx.x * blockDim.x + threadIdx.x;
    const int stride = gridDim.x * blockDim.x;   // multiple of 64 by construction
    const int c      = threadIdx.x & 63;
    float s = 0.f, s2 = 0.f;
    for (int i = idx; i < total; i += stride) {
        const float v = x[i];
        s += v; s2 += v * v;
    }
    __shared__ float sh[256], sh2[256];
    sh[threadIdx.x]  = s;
    sh2[threadIdx.x] = s2;
    __syncthreads();
    if (threadIdx.x < 128) {
        sh[threadIdx.x]  += sh[threadIdx.x + 128];
        sh2[threadIdx.x] += sh2[threadIdx.x + 128];
    }
    __syncthreads();
    if (threadIdx.x < 64) {
        atomicAdd(&stats[c],      sh[threadIdx.x]  + sh[threadIdx.x + 64]);
        atomicAdd(&stats[64 + c], sh2[threadIdx.x] + sh2[threadIdx.x + 64]);
    }
}

__global__ void bn_apply(float* __restrict__ x, const float* __restrict__ stats,
                         const float* __restrict__ gamma, const float* __restrict__ beta,
                         int n, int H)
{
    const int i = blockIdx.x * blockDim.x + threadIdx.x;
    if (i >= n * H) return;
    const int c   = i % H;
    const float inv_n = 1.0f / (float)n;
    const float mean  = stats[c] * inv_n;
    const float var   = stats[64 + c] * inv_n - mean * mean;
    float v = (x[i] - mean) * rsqrtf(var + 1e-5f) * gamma[c] + beta[c];
    x[i] = v > 0.f ? v : 0.01f * v;
}

// ---------------------------------------------------------------------------
// Launcher
// ---------------------------------------------------------------------------
static inline size_t alignup(size_t x) { return (x + 255) & ~(size_t)255; }
static inline int cdiv(int a, int b) { return (a + b - 1) / b; }

extern "C" void kernel_launch(void* const* d_in, const int* in_sizes, int n_in,
                              void* d_out, int out_size, void* d_ws, size_t ws_size,
                              hipStream_t stream)
{
    const int G = 128, H = 64, R = RREL;
    const int N = in_sizes[0] / G;
    const int E = in_sizes[2];
    const int ROWS = 80;                 // N % 80 == 0 for N = 100000

    const float* X     = (const float*)d_in[0];
    const int*   eidx  = (const int*)d_in[1];
    const int*   srcI  = eidx;           // edge_index[0]
    const int*   dstI  = eidx + E;       // edge_index[1]
    const int*   et    = (const int*)d_in[2];
    const float* W1    = (const float*)d_in[3];
    const float* q1    = (const float*)d_in[4];
    const float* k1    = (const float*)d_in[5];
    const float* b1    = (const float*)d_in[6];
    const float* W2    = (const float*)d_in[7];
    const float* q2    = (const float*)d_in[8];
    const float* k2    = (const float*)d_in[9];
    const float* b2    = (const float*)d_in[10];
    const float* gamma = (const float*)d_in[11];
    const float* beta  = (const float*)d_in[12];
    float* out = (float*)d_out;

    // Workspace carve-up
    char* ws = (char*)d_ws;
    size_t o = 0;
    float* XW    = (float*)(ws + o); o = alignup(o + (size_t)R * N * H * 4);
    float* qi    = (float*)(ws + o); o = alignup(o + (size_t)R * N * 4);
    float* kj    = (float*)(ws + o); o = alignup(o + (size_t)R * N * 4);
    float* x2    = (float*)(ws + o); o = alignup(o + (size_t)N * H * 4);
    float* ee    = (float*)(ws + o); o = alignup(o + (size_t)E * 4);
    float* m     = (float*)(ws + o); o = alignup(o + (size_t)N * 4);
    float* denom = (float*)(ws + o); o = alignup(o + (size_t)N * 4);
    float* wq    = (float*)(ws + o); o = alignup(o + (size_t)R * G * 4);
    float* wk    = (float*)(ws + o); o = alignup(o + (size_t)R * G * 4);
    float* stats = (float*)(ws + o); o = alignup(o + 128 * 4);

    // ---------------- Layer 1 (K = G = 128) ----------------
    compute_wqk<<<R, 128, 0, stream>>>(W1, q1, k1, wq, wk, G, H);
    node_qk<<<cdiv(N, 256), 256, 0, stream>>>(X, wq, wk, qi, kj, N, G);
    rgat_gemm_wmma<128><<<dim3(N / ROWS, 1, R), dim3(32, 4), 0, stream>>>(X, W1, XW, N, H);

    fill_f32<<<cdiv(N, 256), 256, 0, stream>>>(m, -INFINITY, N);
    fill_f32<<<cdiv(N, 256), 256, 0, stream>>>(denom, 0.f, N);
    fill_bias<<<cdiv(N * H, 256), 256, 0, stream>>>(x2, b1, N * H, H);

    edge_alpha<<<cdiv(E, 256), 256, 0, stream>>>(srcI, dstI, et, qi, kj, ee, m, E, N);
    edge_exp<<<cdiv(E, 256), 256, 0, stream>>>(dstI, ee, m, denom, E);
    edge_scatter<<<cdiv(E, 4), 256, 0, stream>>>(srcI, dstI, et, ee, denom, XW, x2, E, N, H);
    relu_k<<<cdiv(N * H, 256), 256, 0, stream>>>(x2, N * H);

    // ---------------- Layer 2 (K = H = 64) ----------------
    compute_wqk<<<R, 128, 0, stream>>>(W2, q2, k2, wq, wk, H, H);
    node_qk<<<cdiv(N, 256), 256, 0, stream>>>(x2, wq, wk, qi, kj, N, H);
    rgat_gemm_wmma<64><<<dim3(N / ROWS, 1, R), dim3(32, 4), 0, stream>>>(x2, W2, XW, N, H);

    fill_f32<<<cdiv(N, 256), 256, 0, stream>>>(m, -INFINITY, N);
    fill_f32<<<cdiv(N, 256), 256, 0, stream>>>(denom, 0.f, N);
    fill_bias<<<cdiv(N * H, 256), 256, 0, stream>>>(out, b2, N * H, H);

    edge_alpha<<<cdiv(E, 256), 256, 0, stream>>>(srcI, dstI, et, qi, kj, ee, m, E, N);
    edge_exp<<<cdiv(E, 256), 256, 0, stream>>>(dstI, ee, m, denom, E);
    edge_scatter<<<cdiv(E, 4), 256, 0, stream>>>(srcI, dstI, et, ee, denom, XW, out, E, N, H);

    // ---------------- BatchNorm + leaky_relu(0.01), in place on d_out ------
    fill_f32<<<1, 128, 0, stream>>>(stats, 0.f, 128);
    bn_stats<<<512, 256, 0, stream>>>(out, stats, N * H);
    bn_apply<<<cdiv(N * H, 256), 256, 0, stream>>>(out, stats, gamma, beta, N, H);
}